// Layer_layered2grid_91036126806309
// MI455X (gfx1250) — compile-verified
//
#include <hip/hip_runtime.h>
#include <hip/hip_bf16.h>

// Problem constants derived from the reference:
//   batch B=64, layers L=256, train points N=512, system size NA=514,
//   RHS columns NCOLS = 64*3 = 192, queries M=2048, K padded to 516.
#define NA     514
#define NCAUG  706              // 514 + 192
#define NCOLS  192
#define KPAD   516
#define KSTEPS 129              // KPAD / 4
#define MQ     2048
#define MTILES 128              // MQ / 16
#define NTILES 12               // NCOLS / 16

typedef __attribute__((ext_vector_type(2))) float v2f;
typedef __attribute__((ext_vector_type(8))) float v8f;

// Deterministic train-point positions (softmax over a size-1 axis == 1 =>
// d1 == 700 everywhere; the cumsum is analytic). Select-style: no branches.
__device__ __forceinline__ float posf(int i) {
    float fi = (float)i;
    float p1 = fmaf(700.0f, fi, -1.0f);            // i in 1..255
    float p2 = fmaf(700.0f, fi - 255.0f, 1.0f);    // i in 256..511
    float p  = (i < 256) ? p1 : p2;
    return (i == 0) ? 0.0f : p;
}

// Q matrix entry: phi1(|q - pos_n|) for n<512, then [q, 1], then zero pad.
__device__ __forceinline__ float qval(float q, int n) {
    float v = fmaxf(fabsf(q - posf(n)), 1e-5f);    // sqrt(max(r2, 1e-10))
    v = (n == 512) ? q    : v;
    v = (n == 513) ? 1.0f : v;
    v = (n >= NA)  ? 0.0f : v;                     // K padding
    return v;
}

// ---------------------------------------------------------------------------
// Kernel 1: build augmented matrix Aug[514][706] = [ LHS | RHS ].
// LHS is analytic (batch independent); RHS gathers props from `layered`.
// ---------------------------------------------------------------------------
__global__ __launch_bounds__(256) void build_aug_kernel(
        const float* __restrict__ layered, float* __restrict__ Aug) {
    int idx = blockIdx.x * blockDim.x + threadIdx.x;
    if (idx >= NA * NCAUG) return;
    int i = idx / NCAUG;
    int j = idx - i * NCAUG;
    float val;
    if (j < NA) {
        if (i < 512 && j < 512) {
            val = fmaxf(fabsf(posf(i) - posf(j)), 1e-5f);   // A = phi1(dist)
        } else if (i < 512) {
            val = (j == 512) ? posf(i) : 1.0f;              // B = [pos, 1]
        } else if (j < 512) {
            val = (i == 512) ? posf(j) : 1.0f;              // B^T
        } else {
            val = 0.0f;                                     // 2x2 zero block
        }
    } else {
        int c = j - NA;                 // 0..191
        int b = c / 3;
        int kp = c - 3 * b;
        if (i >= 512) {
            val = 0.0f;                 // spline constraint rows
        } else {
            // props row -> source layer index
            int r;
            if (i == 0)        r = 0;
            else if (i <= 255) r = i - 1;
            else if (i <= 510) r = i - 255;
            else               r = 255;
            val = layered[(b * 256 + r) * 4 + 1 + kp];
        }
    }
    Aug[idx] = val;
}

// ---------------------------------------------------------------------------
// Kernel 2: build Q pre-swizzled into the WMMA A-operand layout.
// Qa[((mt*KSTEPS + s)*32 + lane)*2 + pair] = Q[16*mt + (lane&15)][4*s + (lane>>4)*2 + pair]
// One contiguous b64 per lane per K-step in the GEMM.
// ---------------------------------------------------------------------------
__global__ __launch_bounds__(256) void build_qa_kernel(float* __restrict__ Qa) {
    int idx = blockIdx.x * blockDim.x + threadIdx.x;
    if (idx >= MTILES * KSTEPS * 64) return;
    int pair = idx & 1;
    int lane = (idx >> 1) & 31;
    int s    = (idx >> 6) % KSTEPS;
    int mt   = idx / (KSTEPS * 64);
    int m = 16 * mt + (lane & 15);
    int k = 4 * s + (lane >> 4) * 2 + pair;
    Qa[idx] = qval((float)m * 0.5f, k);
}

// ---------------------------------------------------------------------------
// Kernel 3: in-place Gauss-Jordan with partial pivoting on Aug, single
// workgroup (the system is shared by all batches so we solve it once).
// Epilogue writes X directly in the WMMA B-operand tile layout Xb.
// ---------------------------------------------------------------------------
__global__ __launch_bounds__(1024) void solve_kernel(
        float* __restrict__ Aug, float* __restrict__ Xb) {
    __shared__ float prow[NCAUG];
    __shared__ float sval[1024];
    __shared__ int   sidx[1024];
    __shared__ float srpiv;
    __shared__ int   spiv;
    const int t = threadIdx.x;

    for (int k = 0; k < NA; ++k) {
        // ---- partial pivot search over rows k..513 in column k ----
        float v = -1.0f; int id = k;
        if (t >= k && t < NA) { v = fabsf(Aug[t * NCAUG + k]); id = t; }
        sval[t] = v; sidx[t] = id;
        __syncthreads();
        for (int s = 512; s > 0; s >>= 1) {
            if (t < s) {
                if (sval[t + s] > sval[t]) { sval[t] = sval[t + s]; sidx[t] = sidx[t + s]; }
            }
            __syncthreads();
        }
        if (t == 0) spiv = sidx[0];
        __syncthreads();
        int p = spiv;                       // uniform across workgroup
        if (p != k) {                       // uniform branch
            for (int j = t; j < NCAUG; j += 1024) {
                float a = Aug[k * NCAUG + j];
                float b = Aug[p * NCAUG + j];
                Aug[k * NCAUG + j] = b;
                Aug[p * NCAUG + j] = a;
            }
            __syncthreads();
        }
        // ---- stage pivot row in LDS ----
        for (int j = t; j < NCAUG; j += 1024) prow[j] = Aug[k * NCAUG + j];
        __syncthreads();
        if (t == 0) srpiv = 1.0f / prow[k];
        __syncthreads();
        float rpiv = srpiv;
        // ---- eliminate column k from every other row ----
        if (t < NA && t != k) {
            float f = Aug[t * NCAUG + k] * rpiv;
            if (f != 0.0f) {
                for (int j = k + 1; j < NCAUG; ++j)
                    Aug[t * NCAUG + j] = fmaf(-f, prow[j], Aug[t * NCAUG + j]);
            }
        }
        __syncthreads();
    }

    // ---- Xb[((nt*KSTEPS + s)*32 + lane)*2 + pair] =
    //          X[4*s + (lane>>4)*2 + pair][16*nt + (lane&15)]
    //      with X[r][c] = RHS[r][c] / diag[r], zero for pad rows r >= 514 ----
    for (int idx = t; idx < NTILES * KSTEPS * 64; idx += 1024) {
        int pair = idx & 1;
        int lane = (idx >> 1) & 31;
        int s    = (idx >> 6) % KSTEPS;
        int nt   = idx / (KSTEPS * 64);
        int r = 4 * s + (lane >> 4) * 2 + pair;
        int c = 16 * nt + (lane & 15);
        float x = 0.0f;
        if (r < NA) x = Aug[r * NCAUG + NA + c] / Aug[r * NCAUG + r];
        Xb[idx] = x;
    }
}

// ---------------------------------------------------------------------------
// Kernel 4: Out(2048 x 192) = Q(2048 x 516) * X(516 x 192) via fp32 WMMA.
// Operands are pre-swizzled; inner loop = 2x global_load_b64 + 1x
// v_wmma_f32_16x16x4_f32. One wave per 16x16 tile; grid = (128, 12).
// D layout: VGPR r holds M = r + 8*(lane>>4), N = lane&15.
// ---------------------------------------------------------------------------
__global__ __launch_bounds__(32) void interp_wmma_kernel(
        const float* __restrict__ Qa, const float* __restrict__ Xb,
        float* __restrict__ out) {
    const int lane = threadIdx.x;
    const int mt = blockIdx.x;
    const int nt = blockIdx.y;

    const v2f* __restrict__ aptr =
        (const v2f*)(Qa + ((size_t)mt * KSTEPS * 32 + lane) * 2);
    const v2f* __restrict__ bptr =
        (const v2f*)(Xb + ((size_t)nt * KSTEPS * 32 + lane) * 2);

    v8f acc = {};
    #pragma unroll 3
    for (int s = 0; s < KSTEPS; ++s) {
        v2f a = aptr[s * 32];
        v2f b = bptr[s * 32];
        acc = __builtin_amdgcn_wmma_f32_16x16x4_f32(
                /*neg_a=*/false, a, /*neg_b=*/false, b,
                /*c_mod=*/(short)0, acc, /*reuse_a=*/false, /*reuse_b=*/false);
    }

    // scatter to (64, 2048, 3): column n -> (batch n/3, channel n%3)
    const int n  = nt * 16 + (lane & 15);
    const int bb = n / 3;
    const int kp = n - 3 * bb;
    const int m0 = mt * 16 + (lane >> 4) * 8;
    #pragma unroll
    for (int r = 0; r < 8; ++r) {
        out[(bb * 2048 + m0 + r) * 3 + kp] = acc[r];
    }
}

extern "C" void kernel_launch(void* const* d_in, const int* in_sizes, int n_in,
                              void* d_out, int out_size, void* d_ws, size_t ws_size,
                              hipStream_t stream) {
    const float* layered = (const float*)d_in[0];   // (64, 256, 4) fp32
    float* out = (float*)d_out;                     // (64, 2048, 3) fp32

    float* Aug = (float*)d_ws;                      // 514*706   = 362,884 floats
    float* Xb  = Aug + NA * NCAUG;                  // 12*129*64 =  99,072 floats
    float* Qa  = Xb + NTILES * KSTEPS * 64;         // 128*129*64 = 1,056,768 floats

    int elems = NA * NCAUG;
    build_aug_kernel<<<(elems + 255) / 256, 256, 0, stream>>>(layered, Aug);
    int qelems = MTILES * KSTEPS * 64;
    build_qa_kernel<<<(qelems + 255) / 256, 256, 0, stream>>>(Qa);
    solve_kernel<<<1, 1024, 0, stream>>>(Aug, Xb);
    dim3 grid(MTILES, NTILES);                      // (128, 12)
    interp_wmma_kernel<<<grid, 32, 0, stream>>>(Qa, Xb, out);
}